// LocalCorr_50921132261613
// MI455X (gfx1250) — compile-verified
//
#include <hip/hip_runtime.h>
#include <stdint.h>

#define B_ 4
#define N_ 4
#define C_ 64
#define H_ 128
#define W_ 128
#define HW_ (H_*W_)
#define TW_ 32
#define TH_ 8
#define HLW_ (TW_+2)         // 34
#define HLH_ (TH_+2)         // 10
#define HALO_ (HLW_*HLH_)    // 340
#define EPS_ 1e-12f

typedef __attribute__((ext_vector_type(2))) float v2f;
typedef __attribute__((ext_vector_type(8))) float v8f;

__device__ __forceinline__ int refl(int v, int n) {
  v = (v < 0) ? -v : v;
  v = (v >= n) ? (2*n - 2 - v) : v;
  return v;
}

// ---------------- pass 1: inverse L2 norms over channels ----------------
// ws layout: [0, B*HW)            -> 1/max(||ref(b,:,y,x)||, eps)
//            [B*HW, B*HW+B*N*HW)  -> 1/max(||nbrs(b,i,:,y,x)||, eps)
__global__ __launch_bounds__(256) void norms_kernel(const float* __restrict__ nbrs,
                                                    const float* __restrict__ ref,
                                                    float* __restrict__ ws) {
  int idx = blockIdx.x * 256 + threadIdx.x;
  const int nRef = B_ * HW_;
  if (idx < nRef) {
    int pix = idx & (HW_ - 1);
    int b = idx >> 14;
    const float* p = ref + (size_t)b * C_ * HW_ + pix;
    float s = 0.f;
    #pragma unroll 8
    for (int c = 0; c < C_; ++c) { float v = p[(size_t)c * HW_]; s += v * v; }
    ws[idx] = 1.f / fmaxf(sqrtf(s), EPS_);
  } else {
    int j = idx - nRef;
    if (j < B_ * N_ * HW_) {
      int pix = j & (HW_ - 1);
      int bi = j >> 14;
      const float* p = nbrs + (size_t)bi * C_ * HW_ + pix;
      float s = 0.f;
      #pragma unroll 8
      for (int c = 0; c < C_; ++c) { float v = p[(size_t)c * HW_]; s += v * v; }
      ws[nRef + j] = 1.f / fmaxf(sqrtf(s), EPS_);
    }
  }
}

// ---------------- out[:, N_] = ref (contiguous per-b block) ----------------
__global__ __launch_bounds__(256) void copy_ref_kernel(const float* __restrict__ ref,
                                                       float* __restrict__ out) {
  size_t idx = (size_t)blockIdx.x * 256 + threadIdx.x;   // float4 index
  const size_t perB = (size_t)C_ * HW_ / 4;              // 2^18 float4 per b
  int b = (int)(idx >> 18);
  size_t rem = idx & (perB - 1);
  const float4* src = (const float4*)ref;
  float4* dst = (float4*)out;
  dst[((size_t)(b * (N_ + 1) + N_) << 18) + rem] = src[((size_t)b << 18) + rem];
}

// ---------------- main fused kernel ----------------
__global__ __launch_bounds__(256) void local_corr_kernel(const float* __restrict__ nbrs,
                                                         const float* __restrict__ ref,
                                                         const float* __restrict__ ws,
                                                         float* __restrict__ out) {
  __shared__ float nbr_lds[C_ * HALO_];   // raw, then normalized in place
  __shared__ float invn_lds[HALO_];
  __shared__ float d_lds[9 * TW_ * TH_];

  const int t = threadIdx.x;
  const int x0 = blockIdx.x * TW_;
  const int y0 = blockIdx.y * TH_;
  const int bi = blockIdx.z;             // b*N_ + i
  const int b = bi >> 2, i = bi & 3;

  const float* invRef  = ws;
  const float* invNbrP = ws + B_ * HW_ + (size_t)bi * HW_;
  const float* nbrP    = nbrs + (size_t)bi * C_ * HW_;

  // stage per-pixel inverse norms for the halo tile
  #pragma unroll 1
  for (int h = t; h < HALO_; h += 256) {
    int hy = h / HLW_, hx = h - hy * HLW_;
    int gy = refl(y0 + hy - 1, H_), gx = refl(x0 + hx - 1, W_);
    invn_lds[h] = invNbrP[gy * W_ + gx];
  }

  // async gather raw nbr halo tile (reflect-padded) into LDS: CDNA5 async path
  const uint32_t ldsbase = (uint32_t)(uintptr_t)(&nbr_lds[0]);
  #pragma unroll 1
  for (int e = t; e < C_ * HALO_; e += 256) {   // 21760 = 85*256, no divergence
    int c = e / HALO_, h = e - c * HALO_;
    int hy = h / HLW_, hx = h - hy * HLW_;
    int gy = refl(y0 + hy - 1, H_), gx = refl(x0 + hx - 1, W_);
    uint64_t ga = (uint64_t)(uintptr_t)(nbrP + (size_t)c * HW_ + gy * W_ + gx);
    uint32_t la = ldsbase + (uint32_t)e * 4u;
    asm volatile("global_load_async_to_lds_b32 %0, %1, off" :: "v"(la), "v"(ga) : "memory");
  }
  asm volatile("s_wait_asynccnt 0" ::: "memory");
  __syncthreads();

  // normalize in place -> nbr_lds holds win_n values
  #pragma unroll 1
  for (int e = t; e < C_ * HALO_; e += 256) {
    int h = e % HALO_;
    nbr_lds[e] *= invn_lds[h];
  }
  __syncthreads();

  // ---------- phase A: 9 correlation maps via WMMA f32 16x16x4, diagonal ----------
  const int lane  = t & 31;
  const int wv    = t >> 5;
  const int m     = lane & 15;           // row (A) / col (B) index = pixel in group
  const int khalf = (lane >> 4) << 1;    // K offset 0 or 2 per lane half
  const int sel   = lane & 7;            // diag VGPR index for this lane
  const bool hold = (lane < 8) || (lane >= 24);   // lanes that own a diagonal value
  const int mm    = (lane < 8) ? lane : (lane - 16);  // diag element index (when hold)
  const float* refP = ref + (size_t)b * C_ * HW_;

  #pragma unroll 1
  for (int gi = 0; gi < 2; ++gi) {
    const int g  = wv * 2 + gi;          // 16 groups of 16 pixels, tile-linear
    const int ty = g >> 1;
    const int xh = (g & 1) * 16;
    const int y  = y0 + ty;
    const int x  = x0 + xh + m;
    const float irn = invRef[b * HW_ + y * W_ + x];

    // cache A = ref_n rows for this group (16 K-chunks of 4, 2 regs each)
    float a0[16], a1[16];
    const float* rp = refP + y * W_ + x;
    #pragma unroll
    for (int kk = 0; kk < 16; ++kk) {
      int c0 = kk * 4 + khalf;
      a0[kk] = rp[(size_t)c0 * HW_] * irn;
      a1[kk] = rp[(size_t)(c0 + 1) * HW_] * irn;
    }

    #pragma unroll
    for (int k = 0; k < 9; ++k) {
      const int dy = k / 3 - 1, dx = k % 3 - 1;
      const int hy = ty + 1 + dy;
      const int hx = xh + m + 1 + dx;
      const float* bp = &nbr_lds[hy * HLW_ + hx];
      v8f acc = {0.f, 0.f, 0.f, 0.f, 0.f, 0.f, 0.f, 0.f};
      #pragma unroll
      for (int kk = 0; kk < 16; ++kk) {
        int c0 = kk * 4 + khalf;
        v2f A; A.x = a0[kk];                  A.y = a1[kk];
        v2f Bv; Bv.x = bp[(size_t)c0 * HALO_]; Bv.y = bp[(size_t)(c0 + 1) * HALO_];
        acc = __builtin_amdgcn_wmma_f32_16x16x4_f32(false, A, false, Bv,
                                                    (short)0, acc, false, false);
      }
      // branch-free diagonal gather: lane owns acc[lane&7] if hold
      float dval = acc[0];
      #pragma unroll
      for (int r = 1; r < 8; ++r) dval = (sel == r) ? acc[r] : dval;
      if (hold) d_lds[k * (TW_ * TH_) + g * 16 + mm] = dval;
    }
  }
  __syncthreads();

  // ---------- phase B: softmax over 9 shifts, aggregate, wdiff, store ----------
  {
    const int ty2 = t >> 5, tx2 = t & 31;
    const int y = y0 + ty2, x = x0 + tx2;

    float dv[9], wt[9];
    float dmax = -3.4e38f;
    #pragma unroll
    for (int k = 0; k < 9; ++k) { dv[k] = d_lds[k * (TW_ * TH_) + t]; dmax = fmaxf(dmax, dv[k]); }
    float ssum = 0.f;
    #pragma unroll
    for (int k = 0; k < 9; ++k) { wt[k] = __expf(dv[k] - dmax); ssum += wt[k]; }
    const float isum = 1.f / ssum;
    #pragma unroll
    for (int k = 0; k < 9; ++k) wt[k] *= isum;

    const int hy = ty2 + 1, hx = tx2 + 1;
    const float* base = &nbr_lds[hy * HLW_ + hx];
    const float* np0 = nbrs + (size_t)b * N_ * C_ * HW_ + y * W_ + x;
    float* op = out + (size_t)(b * (N_ + 1) + i) * C_ * HW_ + y * W_ + x;

    #pragma unroll 2
    for (int c = 0; c < C_; ++c) {
      const float* bc = base + (size_t)c * HALO_;
      float agg = 0.f;
      #pragma unroll
      for (int k = 0; k < 9; ++k) {
        int dy = k / 3 - 1, dx = k % 3 - 1;
        agg += wt[k] * bc[dy * HLW_ + dx];
      }
      float s = 0.f, center = 0.f;
      #pragma unroll
      for (int j = 0; j < N_; ++j) {
        float v = np0[(size_t)(j * C_ + c) * HW_];
        s += v;
        if (j == i) center = v;
      }
      float df = center - 0.25f * s;          // nbr - mean
      float wd = __expf(-(df * df));          // exp(ALPHA * diff^2), ALPHA = -1
      op[(size_t)c * HW_] = agg * wd;
    }
  }
}

extern "C" void kernel_launch(void* const* d_in, const int* in_sizes, int n_in,
                              void* d_out, int out_size, void* d_ws, size_t ws_size,
                              hipStream_t stream) {
  const float* nbrs = (const float*)d_in[0];   // (4,4,64,128,128)
  const float* ref  = (const float*)d_in[1];   // (4,64,128,128)
  float* out = (float*)d_out;                  // (4,5,64,128,128)
  float* ws  = (float*)d_ws;                   // >= 327680 floats used

  norms_kernel<<<dim3((B_ * HW_ + B_ * N_ * HW_) / 256), 256, 0, stream>>>(nbrs, ref, ws);
  copy_ref_kernel<<<dim3((B_ * C_ * HW_ / 4) / 256), 256, 0, stream>>>(ref, out);
  dim3 grid(W_ / TW_, H_ / TH_, B_ * N_);
  local_corr_kernel<<<grid, 256, 0, stream>>>(nbrs, ref, ws, out);
}